// SpatialContextEncoder_25967372271646
// MI455X (gfx1250) — compile-verified
//
#include <hip/hip_runtime.h>
#include <hip/hip_bf16.h>
#include <math.h>

#define N_NODES 4096
#define D_MODEL 256
#define N_HEADS 8
#define D_HEAD  32
#define QKV_LD  768
#define ADJ_WPR 128   /* 4096/32 words per adjacency row */

typedef float v2f __attribute__((ext_vector_type(2)));
typedef float v8f __attribute__((ext_vector_type(8)));

// CDNA5 fp32 WMMA: D(16x16,f32) = A(16x4,f32) * B(4x16,f32) + C
__device__ __forceinline__ v8f wmma_f32(v2f a, v2f b, v8f c) {
  return __builtin_amdgcn_wmma_f32_16x16x4_f32(
      /*neg_a=*/false, a, /*neg_b=*/false, b,
      /*c_mod=*/(short)0, c, /*reuse_a=*/false, /*reuse_b=*/false);
}

__global__ void zero_u32(unsigned* __restrict__ p, int n) {
  int i = blockIdx.x * blockDim.x + threadIdx.x;
  if (i < n) p[i] = 0u;
}

__global__ void build_adj(const long long* __restrict__ ei,
                          unsigned* __restrict__ adjw, int E) {
  int i = blockIdx.x * blockDim.x + threadIdx.x;
  if (i >= E) return;
  int s = (int)ei[i];
  int d = (int)ei[E + i];
  if (s == d) return;  // diagonal removed by reference (adj & ~eye)
  atomicOr(&adjw[(size_t)s * ADJ_WPR + (d >> 5)], 1u << (d & 31));
  atomicOr(&adjw[(size_t)d * ADJ_WPR + (s >> 5)], 1u << (s & 31));
}

__global__ void compute_has_nb(const unsigned* __restrict__ adjw,
                               int* __restrict__ has_nb) {
  int r = blockIdx.x * blockDim.x + threadIdx.x;
  if (r >= N_NODES) return;
  const unsigned* w = adjw + (size_t)r * ADJ_WPR;
  unsigned any = 0;
  for (int i = 0; i < ADJ_WPR; ++i) any |= w[i];
  has_nb[r] = any ? 1 : 0;
}

// C[M,N] = A[M,K] @ W[N,K]^T + bias ; one 16x16 tile per wave, fp32 WMMA.
// EPI==1: C = has_nb[row] ? C : emb[row,col]  (out-projection fuse)
template <int EPI>
__global__ void __launch_bounds__(128)
gemm_wmma(const float* __restrict__ A, const float* __restrict__ W,
          const float* __restrict__ bias, float* __restrict__ C,
          int M, int N, int K,
          const float* __restrict__ emb, const int* __restrict__ has_nb) {
  const int lane = threadIdx.x & 31;
  const int wv   = threadIdx.x >> 5;
  const int hl   = lane >> 4;   // half-wave: selects K sub-pair / row block
  const int ln   = lane & 15;
  const int ntiles = N >> 4;
  int tile = blockIdx.x * (blockDim.x >> 5) + wv;
  if (tile >= (M >> 4) * ntiles) return;
  int tm = tile / ntiles, tn = tile % ntiles;

  // A frag: lane<16 holds row=ln K={k,k+1}; lane>=16 holds K={k+2,k+3}
  const float* arow = A + (size_t)(tm * 16 + ln) * K + hl * 2;
  // B frag: B[k][n] = W[n][k]; lane n=ln, VGPR0/1 = K pair
  const float* wrow = W + (size_t)(tn * 16 + ln) * K + hl * 2;

  v8f c = {};
  for (int k = 0; k < K; k += 4) {
    v2f a = *(const v2f*)(arow + k);
    v2f b = *(const v2f*)(wrow + k);
    c = wmma_f32(a, b, c);
  }
  // D layout: VGPR j -> row M=j (lanes<16) / j+8 (lanes>=16), col N=ln
  for (int j = 0; j < 8; ++j) {
    int row = tm * 16 + j + hl * 8;
    int col = tn * 16 + ln;
    float v = c[j] + bias[col];
    if (EPI == 1) v = has_nb[row] ? v : emb[(size_t)row * D_MODEL + col];
    C[(size_t)row * N + col] = v;
  }
}

// Flash-style masked attention: one wave per (16-query tile, head).
__global__ void __launch_bounds__(128)
attn_flash(const float* __restrict__ qkv, const unsigned* __restrict__ adjw,
           float* __restrict__ ctx) {
  __shared__ float plds[4][16][17];  // per-wave P staging (pad avoids bank conflicts)
  const int lane = threadIdx.x & 31;
  const int wv   = threadIdx.x >> 5;
  const int hl   = lane >> 4;
  const int ln   = lane & 15;
  int gw    = blockIdx.x * 4 + wv;
  int qtile = gw >> 3;
  int head  = gw & 7;
  int q0    = qtile * 16;
  int hd0   = head * D_HEAD;
  const float scale = 0.17677669529663689f;  // 1/sqrt(32)

  // Q fragments for all 8 K-steps (DH=32, 4 per WMMA), kept in registers.
  v2f qa[8];
  #pragma unroll
  for (int kk = 0; kk < 8; ++kk)
    qa[kk] = *(const v2f*)(qkv + (size_t)(q0 + ln) * QKV_LD + hd0 + kk * 4 + hl * 2);

  float m[8], l[8];
  v8f acc0 = {}, acc1 = {};
  #pragma unroll
  for (int j = 0; j < 8; ++j) { m[j] = -3.0e38f; l[j] = 0.f; }

  for (int kt = 0; kt < N_NODES / 16; ++kt) {
    int key0 = kt * 16;
    // S = Qh (16x32) * Kh^T (32x16)
    v8f s = {};
    const float* kbase = qkv + (size_t)(key0 + ln) * QKV_LD + 256 + hd0 + hl * 2;
    #pragma unroll
    for (int kk = 0; kk < 8; ++kk) {
      v2f kb = *(const v2f*)(kbase + kk * 4);
      s = wmma_f32(qa[kk], kb, s);
    }
    // Mask + online softmax. Lane slot j covers row (j + hl*8), col (key0+ln).
    int wword  = key0 >> 5;
    int bshift = (key0 & 31) + ln;
    #pragma unroll
    for (int j = 0; j < 8; ++j) {
      int row = q0 + j + hl * 8;
      unsigned wm = adjw[(size_t)row * ADJ_WPR + wword];
      bool bit = (wm >> bshift) & 1u;
      float v = bit ? s[j] * scale : -3.0e38f;
      float mt = v;  // row max across 16 key columns (16-lane halves)
      mt = fmaxf(mt, __shfl_xor(mt, 1, 32));
      mt = fmaxf(mt, __shfl_xor(mt, 2, 32));
      mt = fmaxf(mt, __shfl_xor(mt, 4, 32));
      mt = fmaxf(mt, __shfl_xor(mt, 8, 32));
      float mnew = fmaxf(m[j], mt);
      float corr = expf(m[j] - mnew);
      float p = bit ? expf(v - mnew) : 0.f;  // masked lanes contribute exactly 0
      float ps = p;
      ps += __shfl_xor(ps, 1, 32);
      ps += __shfl_xor(ps, 2, 32);
      ps += __shfl_xor(ps, 4, 32);
      ps += __shfl_xor(ps, 8, 32);
      l[j] = l[j] * corr + ps;
      m[j] = mnew;
      acc0[j] *= corr;
      acc1[j] *= corr;
      plds[wv][j + hl * 8][ln] = p;  // stage P for A-fragment re-layout
    }
    __syncthreads();  // uniform trip count across all 4 waves
    // ctx += P (16x16) * V (16x32): K=16 keys -> 4 WMMA steps, 2 col-chunks
    const float* vbase = qkv + 512 + hd0;
    #pragma unroll
    for (int kk = 0; kk < 4; ++kk) {
      v2f pa;
      pa.x = plds[wv][ln][kk * 4 + hl * 2 + 0];
      pa.y = plds[wv][ln][kk * 4 + hl * 2 + 1];
      int vr = key0 + kk * 4 + hl * 2;
      v2f vb0, vb1;
      vb0.x = vbase[(size_t)vr * QKV_LD + ln];
      vb0.y = vbase[(size_t)(vr + 1) * QKV_LD + ln];
      vb1.x = vbase[(size_t)vr * QKV_LD + 16 + ln];
      vb1.y = vbase[(size_t)(vr + 1) * QKV_LD + 16 + ln];
      acc0 = wmma_f32(pa, vb0, acc0);
      acc1 = wmma_f32(pa, vb1, acc1);
    }
    __syncthreads();
  }
  #pragma unroll
  for (int j = 0; j < 8; ++j) {
    int row = q0 + j + hl * 8;
    float inv = l[j] > 0.f ? 1.f / l[j] : 0.f;  // no-neighbor rows overridden later
    ctx[(size_t)row * D_MODEL + hd0 + ln]      = acc0[j] * inv;
    ctx[(size_t)row * D_MODEL + hd0 + 16 + ln] = acc1[j] * inv;
  }
}

__global__ void __launch_bounds__(256)
ln_gelu(const float* __restrict__ y, const float* __restrict__ g,
        const float* __restrict__ beta, float* __restrict__ out) {
  __shared__ float red[256];
  int row = blockIdx.x, t = threadIdx.x;
  float v = y[(size_t)row * D_MODEL + t];
  red[t] = v;
  __syncthreads();
  for (int s = 128; s > 0; s >>= 1) { if (t < s) red[t] += red[t + s]; __syncthreads(); }
  float mu = red[0] / (float)D_MODEL;
  __syncthreads();
  float d = v - mu;
  red[t] = d * d;
  __syncthreads();
  for (int s = 128; s > 0; s >>= 1) { if (t < s) red[t] += red[t + s]; __syncthreads(); }
  float var = red[0] / (float)D_MODEL;  // biased var, matches jnp.var
  float nv = d * rsqrtf(var + 1e-5f) * g[t] + beta[t];
  float ge = 0.5f * nv * (1.f + erff(nv * 0.7071067811865475f));  // exact GELU
  out[(size_t)row * D_MODEL + t] = ge;
}

extern "C" void kernel_launch(void* const* d_in, const int* in_sizes, int n_in,
                              void* d_out, int out_size, void* d_ws, size_t ws_size,
                              hipStream_t stream) {
  const float*     emb   = (const float*)d_in[0];
  const long long* ei    = (const long long*)d_in[1];  // int64 edge_index [2,E]
  const float*     in_w  = (const float*)d_in[2];
  const float*     in_b  = (const float*)d_in[3];
  const float*     out_w = (const float*)d_in[4];
  const float*     out_b = (const float*)d_in[5];
  const float*     lin_w = (const float*)d_in[6];
  const float*     lin_b = (const float*)d_in[7];
  const float*     ln_g  = (const float*)d_in[8];
  const float*     ln_b2 = (const float*)d_in[9];
  float* out = (float*)d_out;
  const int E = in_sizes[1] / 2;

  char* ws = (char*)d_ws;
  unsigned* adjw = (unsigned*)ws; ws += (size_t)N_NODES * ADJ_WPR * sizeof(unsigned);
  float* qkv = (float*)ws;        ws += (size_t)N_NODES * QKV_LD * sizeof(float);
  float* ctx = (float*)ws;        ws += (size_t)N_NODES * D_MODEL * sizeof(float);
  float* x   = (float*)ws;        ws += (size_t)N_NODES * D_MODEL * sizeof(float);
  float* y   = (float*)ws;        ws += (size_t)N_NODES * D_MODEL * sizeof(float);
  int* hnb   = (int*)ws;

  int adjn = N_NODES * ADJ_WPR;
  zero_u32<<<(adjn + 255) / 256, 256, 0, stream>>>(adjw, adjn);
  build_adj<<<(E + 255) / 256, 256, 0, stream>>>(ei, adjw, E);
  compute_has_nb<<<(N_NODES + 255) / 256, 256, 0, stream>>>(adjw, hnb);

  // qkv = emb @ in_proj_w^T + in_proj_b   [4096,768]
  {
    int tiles = (N_NODES / 16) * (QKV_LD / 16);
    gemm_wmma<0><<<tiles / 4, 128, 0, stream>>>(emb, in_w, in_b, qkv,
                                                N_NODES, QKV_LD, D_MODEL,
                                                nullptr, nullptr);
  }
  // masked flash attention -> ctx [4096,256]
  attn_flash<<<(N_NODES / 16) * N_HEADS / 4, 128, 0, stream>>>(qkv, adjw, ctx);
  // x = where(has_nb, ctx @ out_w^T + out_b, emb)
  {
    int tiles = (N_NODES / 16) * (D_MODEL / 16);
    gemm_wmma<1><<<tiles / 4, 128, 0, stream>>>(ctx, out_w, out_b, x,
                                                N_NODES, D_MODEL, D_MODEL,
                                                emb, hnb);
    // y = x @ lin_w^T + lin_b
    gemm_wmma<0><<<tiles / 4, 128, 0, stream>>>(x, lin_w, lin_b, y,
                                                N_NODES, D_MODEL, D_MODEL,
                                                nullptr, nullptr);
  }
  // LayerNorm + exact GELU -> output
  ln_gelu<<<N_NODES, 256, 0, stream>>>(y, ln_g, ln_b2, out);
}